// GINMalwareClassifier_24137716203810
// MI455X (gfx1250) — compile-verified
//
#include <hip/hip_runtime.h>
#include <hip/hip_bf16.h>

typedef float v2f __attribute__((ext_vector_type(2)));
typedef float v8f __attribute__((ext_vector_type(8)));

#define CH 128
#define PB 128  // partial-reduction blocks for BN stats

// ---------------------------------------------------------------------------
// Edge scatter-add:  AGG[dst] += H[src]   (32 threads per edge, 4ch each)
// ---------------------------------------------------------------------------
__global__ void gin_scatter_add(const float* __restrict__ H,
                                const int* __restrict__ srcidx,
                                const int* __restrict__ dstidx,
                                float* __restrict__ AGG, int nE) {
    long long idx = (long long)blockIdx.x * blockDim.x + threadIdx.x;
    long long total = (long long)nE * 32;
    if (idx >= total) return;
    int e  = (int)(idx >> 5);
    int cq = (int)(idx & 31) << 2;          // channel group of 4
    int s = srcidx[e];
    int d = dstidx[e];
    const float4 v = *(const float4*)(H + (size_t)s * CH + cq);
    float* p = AGG + (size_t)d * CH + cq;
    atomicAdd(p + 0, v.x);
    atomicAdd(p + 1, v.y);
    atomicAdd(p + 2, v.z);
    atomicAdd(p + 3, v.w);
}

// ---------------------------------------------------------------------------
// Fused pre-op + GEMM:  Y[M,128] = preop(A) @ W[128,128] + bias
//   MODE 0: preop = (1+eps)*H + AGG          (GIN combine)
//   MODE 1: preop = relu(A*scale[c]+shift[c]) (BN+ReLU)
// Block = 256 threads = 8 wave32; each wave -> one 16x16 tile via
// 32 chained v_wmma_f32_16x16x4_f32. A tile staged in LDS (padded stride).
// ---------------------------------------------------------------------------
template <int MODE>
__global__ void gin_gemm128(const float* __restrict__ A0,
                            const float* __restrict__ A1,
                            const float* __restrict__ scale,
                            const float* __restrict__ shift,
                            const float* __restrict__ W,
                            const float* __restrict__ bias,
                            float* __restrict__ Y, int M) {
    __shared__ float Atile[16 * 132];       // stride 132 -> conflict-free m-indexed reads
    const int tid  = threadIdx.x;
    const int lane = tid & 31;
    const int wave = tid >> 5;
    const int row0 = blockIdx.x * 16;

    const float epsv = (MODE == 0) ? scale[0] : 0.0f;   // hoisted scalar

    // cooperative load + fused elementwise pre-op
    for (int i = tid; i < 16 * CH; i += 256) {
        int r = i >> 7;
        int c = i & 127;
        int row = row0 + r;
        float v = 0.0f;
        if (row < M) {
            size_t off = (size_t)row * CH + c;
            if (MODE == 0) {
                v = (1.0f + epsv) * A0[off] + A1[off];
            } else {
                float t = A0[off] * scale[c] + shift[c];
                v = t > 0.0f ? t : 0.0f;
            }
        }
        Atile[r * 132 + c] = v;
    }
    __syncthreads();

    const int colb = wave * 16;
    const int m    = lane & 15;
    const int kh   = (lane >> 4) << 1;        // 0 or 2 (K sub-row per lane half)
    const int col  = colb + m;
    const float* Wp = W + col;

    v8f acc = {};
#pragma unroll 8
    for (int k0 = 0; k0 < CH; k0 += 4) {
        v2f a, b;
        a.x = Atile[m * 132 + k0 + kh];
        a.y = Atile[m * 132 + k0 + kh + 1];
        b.x = Wp[(size_t)(k0 + kh) * CH];
        b.y = Wp[(size_t)(k0 + kh + 1) * CH];
        acc = __builtin_amdgcn_wmma_f32_16x16x4_f32(false, a, false, b,
                                                    (short)0, acc, false, false);
    }

    // D layout: VGPR r, lanes 0-15 -> row r ; lanes 16-31 -> row r+8 ; N = lane&15
    const int rbase = row0 + ((lane >> 4) << 3);
    const float bv  = bias[col];
    if (row0 + 16 <= M) {                     // uniform fast path: no per-row guards
#pragma unroll
        for (int r = 0; r < 8; ++r) {
            Y[(size_t)(rbase + r) * CH + col] = acc[r] + bv;
        }
    } else {
#pragma unroll
        for (int r = 0; r < 8; ++r) {
            int row = rbase + r;
            if (row < M) Y[(size_t)row * CH + col] = acc[r] + bv;
        }
    }
}

// ---------------------------------------------------------------------------
// BN statistics: stage 1 -> per-block partial sum/sumsq per channel
// ---------------------------------------------------------------------------
__global__ void bn_stats_partial(const float* __restrict__ Yv,
                                 float* __restrict__ psum,
                                 float* __restrict__ psq, int M) {
    int t = threadIdx.x;                    // channel
    int b = blockIdx.x;
    int per = (M + gridDim.x - 1) / gridDim.x;
    int r0 = b * per;
    int r1 = min(M, r0 + per);
    float s = 0.0f, q = 0.0f;
    for (int r = r0; r < r1; ++r) {
        float v = Yv[(size_t)r * CH + t];
        s += v;
        q += v * v;
    }
    psum[b * CH + t] = s;
    psq[b * CH + t]  = q;
}

// stage 2 -> fold partials into fused scale/shift (deterministic)
__global__ void bn_stats_final(const float* __restrict__ psum,
                               const float* __restrict__ psq,
                               const float* __restrict__ gamma,
                               const float* __restrict__ beta,
                               float* __restrict__ scale,
                               float* __restrict__ shift, int nb, int M) {
    int t = threadIdx.x;
    float s = 0.0f, q = 0.0f;
    for (int b = 0; b < nb; ++b) {
        s += psum[b * CH + t];
        q += psq[b * CH + t];
    }
    float mu  = s / (float)M;
    float var = q / (float)M - mu * mu;
    float inv = rsqrtf(var + 1e-5f);
    float sc  = gamma[t] * inv;
    scale[t] = sc;
    shift[t] = beta[t] - mu * sc;
}

// ---------------------------------------------------------------------------
// Elementwise BN + ReLU (post-conv)
// ---------------------------------------------------------------------------
__global__ void bn_relu_ew(const float* __restrict__ Yv,
                           const float* __restrict__ scale,
                           const float* __restrict__ shift,
                           float* __restrict__ H, long long n) {
    long long i = (long long)blockIdx.x * blockDim.x + threadIdx.x;
    if (i >= n) return;
    int c = (int)(i & 127);
    float v = Yv[i] * scale[c] + shift[c];
    H[i] = v > 0.0f ? v : 0.0f;
}

// ---------------------------------------------------------------------------
// Deterministic per-graph segment sum (batch is sorted): block per graph,
// binary-search the row range, thread per channel.
// ---------------------------------------------------------------------------
__global__ void graph_pool(const float* __restrict__ H,
                           const int* __restrict__ batch,
                           float* __restrict__ psum,
                           int* __restrict__ counts, int M) {
    int g = blockIdx.x;
    int t = threadIdx.x;
    int lo = 0, hi = M;
    while (lo < hi) { int mid = (lo + hi) >> 1; if (batch[mid] < g) lo = mid + 1; else hi = mid; }
    int start = lo;
    lo = start; hi = M;
    while (lo < hi) { int mid = (lo + hi) >> 1; if (batch[mid] < g + 1) lo = mid + 1; else hi = mid; }
    int end = lo;
    float s = 0.0f;
    for (int r = start; r < end; ++r) s += H[(size_t)r * CH + t];
    psum[(size_t)g * CH + t] = s;
    if (t == 0) counts[g] = end - start;
}

// ---------------------------------------------------------------------------
// Classifier head: one block per graph (emb=388 -> 128 relu -> 2)
// ---------------------------------------------------------------------------
__global__ void classifier(const float* __restrict__ pools,   // [3][G][128]
                           const int* __restrict__ counts,
                           const float* __restrict__ gattr,   // [G][4]
                           const float* __restrict__ Wc1,     // [388][128]
                           const float* __restrict__ bc1,
                           const float* __restrict__ Wc2,     // [128][2]
                           const float* __restrict__ bc2,
                           float* __restrict__ out, int G, int GA) {
    __shared__ float emb[392];
    __shared__ float zs[CH];
    int g = blockIdx.x;
    int t = threadIdx.x;
    int clf_in = 3 * CH + GA;                // 388
    float inv = 1.0f / fmaxf((float)counts[g], 1.0f);
    for (int i = t; i < clf_in; i += CH) {
        float v;
        if (i < 3 * CH) {
            int l = i >> 7;
            int c = i & 127;
            v = pools[((size_t)l * G + g) * CH + c] * inv;
        } else {
            v = gattr[(size_t)g * GA + (i - 3 * CH)];
        }
        emb[i] = v;
    }
    __syncthreads();
    float acc = bc1[t];
    for (int k = 0; k < clf_in; ++k) acc += emb[k] * Wc1[(size_t)k * CH + t];
    zs[t] = acc > 0.0f ? acc : 0.0f;
    __syncthreads();
    if (t < 2) {
        float o = bc2[t];
        for (int k = 0; k < CH; ++k) o += zs[k] * Wc2[k * 2 + t];
        out[(size_t)g * 2 + t] = o;
    }
}

// ---------------------------------------------------------------------------
extern "C" void kernel_launch(void* const* d_in, const int* in_sizes, int n_in,
                              void* d_out, int out_size, void* d_ws, size_t ws_size,
                              hipStream_t stream) {
    const float* x     = (const float*)d_in[0];
    const int*   eidx  = (const int*)d_in[1];      // [2, nE]
    const int*   batch = (const int*)d_in[2];
    const float* gattr = (const float*)d_in[3];

    const int M  = in_sizes[0] / CH;               // 100000
    const int nE = in_sizes[1] / 2;                // 1600000
    const int G  = in_sizes[3] / 4;                // 512
    const int GA = 4;

    const int* srcidx = eidx;
    const int* dstidx = eidx + nE;

    // params: 4 + layer*9 : eps,W1,b1,g1,be1,W2,b2,g_out,be_out
    const float* Wc1 = (const float*)d_in[4 + 3 * 9 + 0];
    const float* bc1 = (const float*)d_in[4 + 3 * 9 + 1];
    const float* Wc2 = (const float*)d_in[4 + 3 * 9 + 2];
    const float* bc2 = (const float*)d_in[4 + 3 * 9 + 3];

    // ---- workspace carve-up ----
    const size_t MN = (size_t)M * CH;
    float* AGG   = (float*)d_ws;                   // also reused as Y2
    float* Y1    = AGG + MN;
    float* HBUF  = Y1 + MN;
    float* PSUM  = HBUF + MN;                      // PB*128
    float* PSQ   = PSUM + (size_t)PB * CH;
    float* SCALE = PSQ + (size_t)PB * CH;
    float* SHIFT = SCALE + CH;
    float* POOLS = SHIFT + CH;                     // 3*G*128
    int*   COUNTS = (int*)(POOLS + (size_t)3 * G * CH);

    const dim3 blk256(256), blk128(CH);
    const int gemmBlocks    = (M + 15) / 16;
    const long long scTasks = (long long)nE * 32;
    const int scatterBlocks = (int)((scTasks + 255) / 256);
    const long long ewN     = MN;
    const int ewBlocks      = (int)((ewN + 255) / 256);

    const float* h_in = x;
    for (int l = 0; l < 3; ++l) {
        const float* eps  = (const float*)d_in[4 + l * 9 + 0];
        const float* W1   = (const float*)d_in[4 + l * 9 + 1];
        const float* b1   = (const float*)d_in[4 + l * 9 + 2];
        const float* g1   = (const float*)d_in[4 + l * 9 + 3];
        const float* be1  = (const float*)d_in[4 + l * 9 + 4];
        const float* W2   = (const float*)d_in[4 + l * 9 + 5];
        const float* b2   = (const float*)d_in[4 + l * 9 + 6];
        const float* go   = (const float*)d_in[4 + l * 9 + 7];
        const float* beo  = (const float*)d_in[4 + l * 9 + 8];

        // 1) agg = segment_sum(h[src], dst)
        hipMemsetAsync(AGG, 0, MN * sizeof(float), stream);
        gin_scatter_add<<<scatterBlocks, blk256, 0, stream>>>(h_in, srcidx, dstidx, AGG, nE);

        // 2) Y1 = ((1+eps)*h + agg) @ W1 + b1   (fused, WMMA fp32)
        gin_gemm128<0><<<gemmBlocks, blk256, 0, stream>>>(h_in, AGG, eps, nullptr, W1, b1, Y1, M);

        // 3) BN stats over Y1 -> scale/shift
        bn_stats_partial<<<PB, blk128, 0, stream>>>(Y1, PSUM, PSQ, M);
        bn_stats_final<<<1, blk128, 0, stream>>>(PSUM, PSQ, g1, be1, SCALE, SHIFT, PB, M);

        // 4) Y2 = relu(bn(Y1)) @ W2 + b2   (fused, WMMA fp32; Y2 aliases AGG)
        gin_gemm128<1><<<gemmBlocks, blk256, 0, stream>>>(Y1, nullptr, SCALE, SHIFT, W2, b2, AGG, M);

        // 5) post-conv BN stats
        bn_stats_partial<<<PB, blk128, 0, stream>>>(AGG, PSUM, PSQ, M);
        bn_stats_final<<<1, blk128, 0, stream>>>(PSUM, PSQ, go, beo, SCALE, SHIFT, PB, M);

        // 6) h_out = relu(bn(Y2))   -> HBUF (next layer input)
        bn_relu_ew<<<ewBlocks, blk256, 0, stream>>>(AGG, SCALE, SHIFT, HBUF, ewN);

        // 7) per-graph pooled sums + counts (deterministic, no atomics)
        graph_pool<<<G, blk128, 0, stream>>>(HBUF, batch, POOLS + (size_t)l * G * CH, COUNTS, M);

        h_in = HBUF;
    }

    classifier<<<G, blk128, 0, stream>>>(POOLS, COUNTS, gattr, Wc1, bc1, Wc2, bc2,
                                         (float*)d_out, G, GA);
}